// ForwardWarping_3942779977930
// MI455X (gfx1250) — compile-verified
//
#include <hip/hip_runtime.h>
#include <hip/hip_bf16.h>
#include <stdint.h>

// Problem constants (match reference)
#define NN 8
#define HH 512
#define WW 1024
#define HWSZ (HH * WW)
#define WQ (WW / 4)
#define BIGF 1.0e8f
#define INVALIDF 1.0e6f
#define EMPTY_SRC 0xffffffffu
#define CHUNK 2048  // pixels per 256-thread block in the projection kernel

typedef float v2f __attribute__((ext_vector_type(2)));
typedef float v8f __attribute__((ext_vector_type(8)));

// Order-preserving map float -> uint32 (monotonic for all finite floats)
__device__ __forceinline__ uint32_t enc_f32(float f) {
  uint32_t u = __float_as_uint(f);
  return (u & 0x80000000u) ? ~u : (u | 0x80000000u);
}
__device__ __forceinline__ float dec_f32(uint32_t m) {
  uint32_t u = (m & 0x80000000u) ? (m & 0x7fffffffu) : ~m;
  return __uint_as_float(u);
}

// ---------------------------------------------------------------------------
// Kernel 1: initialize z-buffer keys to (enc(BIG) << 32) | EMPTY_SRC
// ---------------------------------------------------------------------------
__global__ void fw_init_keys(unsigned long long* __restrict__ keys) {
  int gid = blockIdx.x * blockDim.x + threadIdx.x;
  if (gid < NN * HWSZ) {
    unsigned long long sent =
        ((unsigned long long)enc_f32(BIGF) << 32) | (unsigned long long)EMPTY_SRC;
    keys[gid] = sent;
  }
}

// ---------------------------------------------------------------------------
// Kernel 2: projection via V_WMMA_F32_16X16X4_F32 + atomic-min z-buffer splat.
// Each wave processes 16 pixels per WMMA: D = Q(16x4, rows 0..3 live) x V(4x16)
//   Q row 0..2 = K3*(R*K^-1 | t) + [0|Kcol3]  -> proj x,y,z
//   Q row 3    = (R*K^-1) row2 | t.z          -> dist (camera z)
//   V column p = [u_p*d_p, v_p*d_p, d_p, 1]
// Lane p (<16) receives proj_x,proj_y,proj_z,dist in acc[0..3].
// ---------------------------------------------------------------------------
__global__ __launch_bounds__(256) void fw_project_scatter(
    const float* __restrict__ depth, const float* __restrict__ T,
    const float* __restrict__ Km, unsigned long long* __restrict__ keys) {
  __shared__ float qsh[16];

  const int blocksPerImage = HWSZ / CHUNK;  // 256
  const int n = blockIdx.x / blocksPerImage;
  const int p0 = (blockIdx.x % blocksPerImage) * CHUNK;

  if (threadIdx.x == 0) {
    // K (row-major 4x4)
    float k00 = Km[0], k01 = Km[1], k02 = Km[2], k03 = Km[3];
    float k10 = Km[4], k11 = Km[5], k12 = Km[6], k13 = Km[7];
    float k20 = Km[8], k21 = Km[9], k22 = Km[10], k23 = Km[11];
    // inv(K[:3,:3]) by adjugate
    float det = k00 * (k11 * k22 - k12 * k21) - k01 * (k10 * k22 - k12 * k20) +
                k02 * (k10 * k21 - k11 * k20);
    float id = 1.0f / det;
    float i00 = (k11 * k22 - k12 * k21) * id;
    float i01 = -(k01 * k22 - k02 * k21) * id;
    float i02 = (k01 * k12 - k02 * k11) * id;
    float i10 = -(k10 * k22 - k12 * k20) * id;
    float i11 = (k00 * k22 - k02 * k20) * id;
    float i12 = -(k00 * k12 - k02 * k10) * id;
    float i20 = (k10 * k21 - k11 * k20) * id;
    float i21 = -(k00 * k21 - k01 * k20) * id;
    float i22 = (k00 * k11 - k01 * k10) * id;
    // T[n] (row-major 4x4); R = T[:3,:3], t = T[:3,3]
    const float* Tn = T + n * 16;
    float r00 = Tn[0], r01 = Tn[1], r02 = Tn[2], t0 = Tn[3];
    float r10 = Tn[4], r11 = Tn[5], r12 = Tn[6], t1 = Tn[7];
    float r20 = Tn[8], r21 = Tn[9], r22 = Tn[10], t2 = Tn[11];
    // M = R * invK3
    float m00 = r00 * i00 + r01 * i10 + r02 * i20;
    float m01 = r00 * i01 + r01 * i11 + r02 * i21;
    float m02 = r00 * i02 + r01 * i12 + r02 * i22;
    float m10 = r10 * i00 + r11 * i10 + r12 * i20;
    float m11 = r10 * i01 + r11 * i11 + r12 * i21;
    float m12 = r10 * i02 + r11 * i12 + r12 * i22;
    float m20 = r20 * i00 + r21 * i10 + r22 * i20;
    float m21 = r20 * i01 + r21 * i11 + r22 * i21;
    float m22 = r20 * i02 + r21 * i12 + r22 * i22;
    // P = K3 * M ;  Pt = K3 * t + K[:3,3]
    qsh[0]  = k00 * m00 + k01 * m10 + k02 * m20;
    qsh[1]  = k00 * m01 + k01 * m11 + k02 * m21;
    qsh[2]  = k00 * m02 + k01 * m12 + k02 * m22;
    qsh[3]  = k00 * t0 + k01 * t1 + k02 * t2 + k03;
    qsh[4]  = k10 * m00 + k11 * m10 + k12 * m20;
    qsh[5]  = k10 * m01 + k11 * m11 + k12 * m21;
    qsh[6]  = k10 * m02 + k11 * m12 + k12 * m22;
    qsh[7]  = k10 * t0 + k11 * t1 + k12 * t2 + k13;
    qsh[8]  = k20 * m00 + k21 * m10 + k22 * m20;
    qsh[9]  = k20 * m01 + k21 * m11 + k22 * m21;
    qsh[10] = k20 * m02 + k21 * m12 + k22 * m22;
    qsh[11] = k20 * t0 + k21 * t1 + k22 * t2 + k23;
    qsh[12] = m20;  qsh[13] = m21;  qsh[14] = m22;  qsh[15] = t2;
  }
  __syncthreads();

  const int lane = threadIdx.x & 31;
  const int wave = threadIdx.x >> 5;
  const int r = lane & 15;
  const bool hi = lane >= 16;

  // A matrix (16x4 f32): lane L<16 holds {Q[L][0], Q[L][1]}, lane L>=16 holds
  // {Q[L-16][2], Q[L-16][3]}; rows >= 4 are zero-padded.
  v2f amat;
  amat.x = 0.0f;
  amat.y = 0.0f;
  if (r < 4) {
    amat.x = qsh[r * 4 + (hi ? 2 : 0)];
    amat.y = qsh[r * 4 + (hi ? 3 : 1)];
  }

  const float* dptr = depth + (size_t)n * HWSZ;
  unsigned long long* kptr = keys + (size_t)n * HWSZ;

  const int wbase = p0 + wave * (CHUNK / 8);  // 256 pixels per wave
  #pragma unroll 2
  for (int it = 0; it < (CHUNK / 8) / 16; ++it) {
    int pix = wbase + it * 16 + r;
    float d = dptr[pix];
    // gfx1250 streaming prefetch for the next chunk of the depth stream
    __builtin_prefetch(dptr + pix + 64, 0, 1);

    float u = (float)(pix & (WW - 1));
    float v = (float)(pix >> 10);

    // B matrix (4x16 f32): lanes 0-15 = rows K=0,1 -> (u*d, v*d);
    //                      lanes 16-31 = rows K=2,3 -> (d, 1).
    v2f bmat;
    if (!hi) {
      bmat.x = u * d;
      bmat.y = v * d;
    } else {
      bmat.x = d;
      bmat.y = 1.0f;
    }

    v8f acc = {};
    acc = __builtin_amdgcn_wmma_f32_16x16x4_f32(
        /*neg_a=*/false, amat, /*neg_b=*/false, bmat,
        /*c_mod=*/(short)0, acc, /*reuse_a=*/false, /*reuse_b=*/false);

    if (!hi) {
      float px = acc[0], py = acc[1], pz = acc[2], dist = acc[3];
      if (dist < INVALIDF) {
        float zd = pz + 1e-7f;
        float uo = rintf(px / zd);
        float vo = rintf(py / zd);
        uo = fminf(fmaxf(uo, 0.0f), (float)(WW - 1));
        vo = fminf(fmaxf(vo, 0.0f), (float)(HH - 1));
        int lin = (int)vo * WW + (int)uo;
        // min over (depth, src_index): min depth wins; ties -> smaller index,
        // exactly matching the reference's stable-sort splat order.
        unsigned long long key =
            ((unsigned long long)enc_f32(dist) << 32) | (unsigned long long)(uint32_t)pix;
        atomicMin(kptr + lin, key);
      }
    }
  }
}

// ---------------------------------------------------------------------------
// Kernel 3: resolve. For middle columns [W/4, 3W/4), take argmin-depth over
// {self, up, left, up-left} (clamped inside the slice, first-occurrence ties),
// then gather RGB from the winning source pixel and emit img/depth/mask.
// ---------------------------------------------------------------------------
__global__ __launch_bounds__(256) void fw_resolve(
    const float* __restrict__ img, const unsigned long long* __restrict__ keys,
    float* __restrict__ out) {
  int gid = blockIdx.x * blockDim.x + threadIdx.x;
  if (gid >= NN * HWSZ) return;
  int n = gid / HWSZ;
  int rem = gid - n * HWSZ;
  int i = rem >> 10;        // row
  int j = rem & (WW - 1);   // col

  const unsigned long long* kp = keys + (size_t)n * HWSZ;
  unsigned long long kw;
  if (j >= WQ && j < 3 * WQ) {
    int jr = j - WQ;
    int i1 = (i == 0) ? i : i - 1;          // s01: shift down (row-1, clamped)
    int j1 = (jr == 0) ? j : j - 1;         // s10: shift right (col-1, clamped to slice)
    bool dd = (i >= 1) && (jr >= 1);        // s11: diagonal only when both in range
    int i3 = dd ? i - 1 : i;
    int j3 = dd ? j - 1 : j;
    unsigned long long k0 = kp[i * WW + j];
    unsigned long long k1 = kp[i1 * WW + j];
    unsigned long long k2 = kp[i * WW + j1];
    unsigned long long k3 = kp[i3 * WW + j3];
    float d0 = dec_f32((uint32_t)(k0 >> 32));
    float d1 = dec_f32((uint32_t)(k1 >> 32));
    float d2 = dec_f32((uint32_t)(k2 >> 32));
    float d3 = dec_f32((uint32_t)(k3 >> 32));
    kw = k0;
    float dw = d0;
    if (d1 < dw) { dw = d1; kw = k1; }
    if (d2 < dw) { dw = d2; kw = k2; }
    if (d3 < dw) { dw = d3; kw = k3; }
  } else {
    kw = kp[rem];
  }

  uint32_t src = (uint32_t)kw;
  float dep = dec_f32((uint32_t)(kw >> 32));
  float cr = 0.0f, cg = 0.0f, cb = 0.0f;
  if (src != EMPTY_SRC) {
    const float* ib = img + (size_t)n * 3 * HWSZ + src;
    cr = ib[0];
    cg = ib[HWSZ];
    cb = ib[2 * HWSZ];
  }
  // mask = ((dep==0) + (dep>INVALID)) > 0
  float mask = ((dep == 0.0f) || (dep > INVALIDF)) ? 1.0f : 0.0f;

  size_t px = rem;
  out[((size_t)n * 3 + 0) * HWSZ + px] = cr;
  out[((size_t)n * 3 + 1) * HWSZ + px] = cg;
  out[((size_t)n * 3 + 2) * HWSZ + px] = cb;
  out[(size_t)NN * 3 * HWSZ + (size_t)n * HWSZ + px] = dep;
  out[(size_t)NN * 4 * HWSZ + (size_t)n * HWSZ + px] = mask;
}

// ---------------------------------------------------------------------------
// Launcher. Inputs: img (N,3,H,W) f32, depth (N,1,H,W) f32, T (N,4,4) f32,
// K (4,4) f32. Output: [nv_img (N,3,H,W) | nv_depth (N,1,H,W) | nv_mask] f32.
// Workspace: N*H*W u64 z-buffer keys (32 MB).
// ---------------------------------------------------------------------------
extern "C" void kernel_launch(void* const* d_in, const int* in_sizes, int n_in,
                              void* d_out, int out_size, void* d_ws, size_t ws_size,
                              hipStream_t stream) {
  const float* img = (const float*)d_in[0];
  const float* depth = (const float*)d_in[1];
  const float* T = (const float*)d_in[2];
  const float* Km = (const float*)d_in[3];
  float* out = (float*)d_out;
  unsigned long long* keys = (unsigned long long*)d_ws;

  const int total = NN * HWSZ;
  fw_init_keys<<<(total + 255) / 256, 256, 0, stream>>>(keys);
  fw_project_scatter<<<NN * (HWSZ / CHUNK), 256, 0, stream>>>(depth, T, Km, keys);
  fw_resolve<<<(total + 255) / 256, 256, 0, stream>>>(img, keys, out);
}